// GNN_Cluster_17686675325067
// MI455X (gfx1250) — compile-verified
//
#include <hip/hip_runtime.h>
#include <math.h>

// ---------------------------------------------------------------------------
// MI455X (gfx1250) implementation of the GNN clustering pipeline.
// All heavy math runs through V_WMMA_F32_16X16X4_F32 (fp32 WMMA), wave32.
// ---------------------------------------------------------------------------

typedef __attribute__((ext_vector_type(2))) float v2f;
typedef __attribute__((ext_vector_type(8))) float v8f;

#define NMAT 1024
#define HDIM 128

static __device__ __forceinline__ v8f wmma4(v2f a, v2f b, v8f c) {
  // D = A(16x4) * B(4x16) + C(16x16), fp32.
  return __builtin_amdgcn_wmma_f32_16x16x4_f32(
      /*neg_a=*/false, a, /*neg_b=*/false, b,
      /*c_mod=*/(short)0, c, /*reuse_a=*/false, /*reuse_b=*/false);
}

// ---------------------------------------------------------------------------
// prep: notew[i][h] = note[i][h]*w[h]; noteT[h][i] = note[i][h];
//       u[i] = sum_h w[h]*note[i][h]^2
// grid = 1024 blocks x 128 threads (one block per row i)
// ---------------------------------------------------------------------------
__global__ __launch_bounds__(HDIM) void prep_kernel(
    const float* __restrict__ note, const float* __restrict__ w,
    float* __restrict__ notew, float* __restrict__ noteT, float* __restrict__ u) {
  const int i = blockIdx.x;
  const int h = threadIdx.x;
  const float v  = note[i * HDIM + h];
  const float wh = w[h];
  notew[i * HDIM + h] = v * wh;
  noteT[h * NMAT + i] = v;
  __shared__ float red[HDIM];
  red[h] = wh * v * v;
  __syncthreads();
  for (int s = HDIM / 2; s > 0; s >>= 1) {
    if (h < s) red[h] += red[h + s];
    __syncthreads();
  }
  if (h == 0) u[i] = red[0];
}

// ---------------------------------------------------------------------------
// Generic fp32 WMMA GEMM: C = epilogue(A[MxK] @ B[KxN])
//   mode 0: C = acc
//   mode 1: C = 1.5*I - 0.5*acc              (Newton-Schulz T matrix)
//   mode 2: C = sigmoid(u[r]+u[c]-2*acc+b)   (grouping) + Frobenius^2 partials
// Block: 128 threads = 4 waves stacked in M; wave tile 32x64 (2x4 WMMA accs
// = 8 independent v_wmma per K-step for back-to-back XDL issue);
// block tile 128x64. Requires M%128==0, N%64==0, K%4==0 (true for all calls).
// ---------------------------------------------------------------------------
__global__ __launch_bounds__(128) void gemm_wmma(
    const float* __restrict__ A, const float* __restrict__ B, float* __restrict__ C,
    int M, int N, int K, int lda, int ldb, int ldc, int mode,
    const float* __restrict__ u, const float* __restrict__ biasPtr,
    float* __restrict__ froPart) {
  (void)M; (void)N;
  const int tid   = threadIdx.x;
  const int wave  = tid >> 5;    // 0..3 (wave-rows)
  const int lane  = tid & 31;
  const int half  = lane >> 4;   // 0: lanes 0-15, 1: lanes 16-31
  const int lm    = lane & 15;
  const int rowBase = blockIdx.y * 128 + wave * 32;
  const int colBase = blockIdx.x * 64;

  v8f acc[2][4];
  for (int i = 0; i < 2; ++i)
    for (int j = 0; j < 4; ++j)
      for (int v = 0; v < 8; ++v) acc[i][j][v] = 0.0f;

  const float* Arow0 = A + (size_t)(rowBase + lm) * lda;       // wave row-tile 0
  const float* Arow1 = A + (size_t)(rowBase + 16 + lm) * lda;  // wave row-tile 1

  for (int k = 0; k < K; k += 4) {
    const int ka = k + 2 * half;  // lanes 0-15: K=k,k+1  lanes 16-31: K=k+2,k+3
    // A 16x4 fragments (ISA 7.12.2: VGPR0 = first K of pair, VGPR1 = second)
    v2f a0, a1;
    a0[0] = Arow0[ka];  a0[1] = Arow0[ka + 1];
    a1[0] = Arow1[ka];  a1[1] = Arow1[ka + 1];
    // B 4x16 fragments (row of 16 N-values striped across 16 lanes)
    const float* Bk0 = B + (size_t)ka * ldb + colBase + lm;
    const float* Bk1 = Bk0 + ldb;
    v2f b[4];
    for (int j = 0; j < 4; ++j) {
      b[j][0] = Bk0[j * 16];
      b[j][1] = Bk1[j * 16];
    }
    // Prefetch 32 K-steps ahead (speculative: dropped in HW if past mapping)
    __builtin_prefetch(Arow0 + ka + 128, 0, 3);
    __builtin_prefetch(Bk0 + (size_t)128 * ldb, 0, 3);

    for (int j = 0; j < 4; ++j) {
      acc[0][j] = wmma4(a0, b[j], acc[0][j]);
      acc[1][j] = wmma4(a1, b[j], acc[1][j]);
    }
  }

  // Epilogue. C/D layout: VGPR v -> M = v + 8*half, N = lane&15.
  float lsum = 0.0f;
  const float bias = (mode == 2) ? *biasPtr : 0.0f;
  for (int ti = 0; ti < 2; ++ti) {
    for (int tj = 0; tj < 4; ++tj) {
      const int r0 = rowBase + ti * 16 + 8 * half;
      const int c  = colBase + tj * 16 + lm;
      for (int v = 0; v < 8; ++v) {
        const int r = r0 + v;
        float x = acc[ti][tj][v];
        if (mode == 1) {
          x = (r == c ? 1.5f : 0.0f) - 0.5f * x;
        } else if (mode == 2) {
          x = u[r] + u[c] - 2.0f * x + bias;
          x = 1.0f / (1.0f + expf(-x));
          lsum += x * x;
        }
        C[(size_t)r * ldc + c] = x;
      }
    }
  }

  if (mode == 2) {
    // Deterministic per-block Frobenius^2 partial (no float atomics).
    __shared__ float red[128];
    red[tid] = lsum;
    __syncthreads();
    for (int s = 64; s > 0; s >>= 1) {
      if (tid < s) red[tid] += red[tid + s];
      __syncthreads();
    }
    if (tid == 0) froPart[blockIdx.y * gridDim.x + blockIdx.x] = red[0];
  }
}

// ---------------------------------------------------------------------------
// Fixed-order reduction of the 128 Frobenius partials -> fro2 scalar.
// ---------------------------------------------------------------------------
__global__ __launch_bounds__(128) void fro_reduce_kernel(
    const float* __restrict__ part, float* __restrict__ fro2) {
  __shared__ float red[128];
  const int t = threadIdx.x;
  red[t] = part[t];
  __syncthreads();
  for (int s = 64; s > 0; s >>= 1) {
    if (t < s) red[t] += red[t + s];
    __syncthreads();
  }
  if (t == 0) *fro2 = red[0];
}

// ---------------------------------------------------------------------------
// Y0 = A / ||A||_F ;  Z0 = I     (fro2 holds ||A||_F^2)
// ---------------------------------------------------------------------------
__global__ __launch_bounds__(256) void yz_init_kernel(
    const float* __restrict__ A, const float* __restrict__ fro2,
    float* __restrict__ Y, float* __restrict__ Z) {
  const int e = blockIdx.x * 256 + threadIdx.x;
  const float inv = rsqrtf(*fro2);  // 1/normA
  Y[e] = A[e] * inv;
  const int r = e >> 10, c = e & (NMAT - 1);
  Z[e] = (r == c) ? 1.0f : 0.0f;
}

// ---------------------------------------------------------------------------
// clustering = softmax_row( Y * sqrt(normA) );  normA = sqrt(fro2)
// one 256-thread block per row; 4 columns per thread.
// ---------------------------------------------------------------------------
__global__ __launch_bounds__(256) void row_softmax_kernel(
    const float* __restrict__ Y, const float* __restrict__ fro2,
    float* __restrict__ out) {
  const int row = blockIdx.x;
  const int tid = threadIdx.x;
  const float scale = sqrtf(sqrtf(*fro2));  // sqrt(normA) = fro2^(1/4)
  const float* yr = Y + (size_t)row * NMAT;
  float x[4];
  float m = -3.402823466e38f;
  for (int t = 0; t < 4; ++t) {
    x[t] = yr[tid + t * 256] * scale;
    m = fmaxf(m, x[t]);
  }
  __shared__ float red[256];
  red[tid] = m;
  __syncthreads();
  for (int s = 128; s > 0; s >>= 1) {
    if (tid < s) red[tid] = fmaxf(red[tid], red[tid + s]);
    __syncthreads();
  }
  const float rowmax = red[0];
  __syncthreads();
  float lsum = 0.0f;
  for (int t = 0; t < 4; ++t) {
    x[t] = expf(x[t] - rowmax);
    lsum += x[t];
  }
  red[tid] = lsum;
  __syncthreads();
  for (int s = 128; s > 0; s >>= 1) {
    if (tid < s) red[tid] += red[tid + s];
    __syncthreads();
  }
  const float inv = 1.0f / red[0];
  float* orow = out + (size_t)row * NMAT;
  for (int t = 0; t < 4; ++t) orow[tid + t * 256] = x[t] * inv;
}

// ---------------------------------------------------------------------------
// Host-side orchestration
// ---------------------------------------------------------------------------
static inline void launch_gemm(const float* A, const float* B, float* C,
                               int M, int N, int K, int lda, int ldb, int ldc,
                               int mode, const float* u, const float* bias,
                               float* part, hipStream_t stream) {
  dim3 grid(N / 64, M / 128);
  gemm_wmma<<<grid, dim3(128), 0, stream>>>(A, B, C, M, N, K, lda, ldb, ldc,
                                            mode, u, bias, part);
}

extern "C" void kernel_launch(void* const* d_in, const int* in_sizes, int n_in,
                              void* d_out, int out_size, void* d_ws, size_t ws_size,
                              hipStream_t stream) {
  (void)in_sizes; (void)n_in; (void)out_size; (void)ws_size;
  const float* note = (const float*)d_in[0];   // [1024,128]
  const float* adj  = (const float*)d_in[1];   // [1024,1024]
  const float* w    = (const float*)d_in[2];   // [128]
  const float* bptr = (const float*)d_in[3];   // scalar

  float* out = (float*)d_out;
  float* note_out = out;                             // [1024,128]
  float* adj_out  = out + NMAT * HDIM;               // [1024,1024]
  float* clu      = out + NMAT * HDIM + NMAT * NMAT; // [1024,1024]

  // Workspace layout (floats)
  const size_t MAT = (size_t)NMAT * NMAT;
  float* ws   = (float*)d_ws;
  float* R0   = ws + 0 * MAT;  // grouping A, then T each NS iteration
  float* Yb[2] = { ws + 1 * MAT, ws + 2 * MAT };
  float* Zb[2] = { ws + 3 * MAT, ws + 4 * MAT };
  float* notew = ws + 5 * MAT;                 // [1024,128]
  float* noteT = notew + NMAT * HDIM;          // [128,1024]
  float* u     = noteT + HDIM * NMAT;          // [1024]
  float* part  = u + NMAT;                     // [128] fro partials
  float* fro2  = part + 128;                   // scalar

  // 1) prep: notew, noteT, u
  prep_kernel<<<dim3(NMAT), dim3(HDIM), 0, stream>>>(note, w, notew, noteT, u);

  // 2) grouping = sigmoid(u_i + u_j - 2*(notew @ noteT) + b); fro partials
  launch_gemm(notew, noteT, R0, NMAT, NMAT, HDIM, HDIM, NMAT, NMAT,
              /*mode=*/2, u, bptr, part, stream);
  fro_reduce_kernel<<<dim3(1), dim3(128), 0, stream>>>(part, fro2);

  // 3) Y0 = A/normA ; Z0 = I
  yz_init_kernel<<<dim3(MAT / 256), dim3(256), 0, stream>>>(R0, fro2, Yb[0], Zb[0]);

  // 4) 12 Newton-Schulz iterations (deterministic ping-pong)
  int cur = 0;
  for (int it = 0; it < 12; ++it) {
    // T = 1.5*I - 0.5*(Z @ Y)
    launch_gemm(Zb[cur], Yb[cur], R0, NMAT, NMAT, NMAT, NMAT, NMAT, NMAT,
                /*mode=*/1, nullptr, nullptr, nullptr, stream);
    // Ynew = Y @ T
    launch_gemm(Yb[cur], R0, Yb[1 - cur], NMAT, NMAT, NMAT, NMAT, NMAT, NMAT,
                0, nullptr, nullptr, nullptr, stream);
    // Znew = T @ Z
    launch_gemm(R0, Zb[cur], Zb[1 - cur], NMAT, NMAT, NMAT, NMAT, NMAT, NMAT,
                0, nullptr, nullptr, nullptr, stream);
    cur ^= 1;
  }
  // after 12 iterations cur == 0 -> final Y in Yb[0]

  // 5) clustering = softmax_row(Y * sqrt(normA))  -> directly into d_out slot
  row_softmax_kernel<<<dim3(NMAT), dim3(256), 0, stream>>>(Yb[0], fro2, clu);

  // 6) note_out = clustering @ note   [1024x128]
  launch_gemm(clu, note, note_out, NMAT, HDIM, NMAT, NMAT, HDIM, HDIM,
              0, nullptr, nullptr, nullptr, stream);

  // 7) adj_out = (clustering @ adj) @ clustering
  float* tmp = Zb[1];  // Z buffers dead after NS
  launch_gemm(clu, adj, tmp, NMAT, NMAT, NMAT, NMAT, NMAT, NMAT,
              0, nullptr, nullptr, nullptr, stream);
  launch_gemm(tmp, clu, adj_out, NMAT, NMAT, NMAT, NMAT, NMAT, NMAT,
              0, nullptr, nullptr, nullptr, stream);
}